// BERT_CRF_NER_17214228922988
// MI455X (gfx1250) — compile-verified
//
#include <hip/hip_runtime.h>
#include <math.h>

typedef __attribute__((ext_vector_type(2))) float v2f;
typedef __attribute__((ext_vector_type(8))) float v8f;

#define BB    512
#define TT    128
#define HH    768
#define LL    12
#define STARTL 9
#define NEGV  (-10000.0f)
#define SWS   772   // padded LDS row stride (768 % 64 == 0 -> bank conflicts; 772 % 64 == 4)

// ---------------------------------------------------------------------------
// Kernel 1: feats[b*T+t, l] = dot(X[b*T+t, :], W[l, :]) + bias[l]
// One wave32 per 16-row M tile, f32 WMMA 16x16x4, K swept 768/4 = 192 steps.
// W (12x768) zero-padded to 16 rows, staged in LDS, shared by 4 waves/block.
// ---------------------------------------------------------------------------
__global__ __launch_bounds__(128) void feats_gemm_kernel(
    const float* __restrict__ X, const float* __restrict__ W,
    const float* __restrict__ bias, float* __restrict__ feats) {
  __shared__ float sW[16 * SWS];
  const int tid = threadIdx.x;

  // Cooperative, coalesced load of padded W into LDS.
  for (int i = tid; i < 16 * HH; i += 128) {
    int row = i / HH;
    int col = i - row * HH;
    sW[row * SWS + col] = (row < LL) ? W[row * HH + col] : 0.0f;
  }
  __syncthreads();

  const int wave = tid >> 5;
  const int lane = tid & 31;
  const int half = lane >> 4;   // 0 -> K = k+0,k+1 ; 1 -> K = k+2,k+3
  const int idx16 = lane & 15;  // A: M row ; B: N column (they coincide here)

  const long row0 = (long)blockIdx.x * 64 + (long)wave * 16;
  const float* aptr = X + (row0 + idx16) * HH + half * 2;
  const float* bptr = sW + idx16 * SWS + half * 2;

  v8f c = {};
#pragma unroll 4
  for (int k = 0; k < HH; k += 4) {
    v2f a  = *(const v2f*)(aptr + k);
    v2f bf = *(const v2f*)(bptr + k);
    // D = A(16x4,f32) * B(4x16,f32) + C ; exact f32 accumulation.
    c = __builtin_amdgcn_wmma_f32_16x16x4_f32(false, a, false, bf,
                                              (short)0, c, false, false);
  }

  // C/D layout: VGPR r -> lanes 0-15: (M=r, N=lane); lanes 16-31: (M=8+r, N=lane-16)
  const int col = lane & 15;
  if (col < LL) {
    const float bv = bias[col];
#pragma unroll
    for (int r = 0; r < 8; ++r) {
      long m = row0 + half * 8 + r;
      feats[m * LL + col] = c[r] + bv;
    }
  }
}

// ---------------------------------------------------------------------------
// Kernel 2: CRF forward recursion + gold score, one wave32 per batch element.
// Lane j (j<12) owns alpha[j]; alpha exchanged via wave32 shuffles.
// ---------------------------------------------------------------------------
__global__ __launch_bounds__(32) void crf_kernel(
    const float* __restrict__ feats, const float* __restrict__ trans,
    const int* __restrict__ labels, float* __restrict__ per_batch) {
  const int b    = blockIdx.x;
  const int lane = threadIdx.x;
  const int j    = (lane < LL) ? lane : 0;  // clamp to avoid OOB on idle lanes

  float tr[LL];
#pragma unroll
  for (int i = 0; i < LL; ++i) tr[i] = trans[j * LL + i];

  float alpha = (lane == STARTL) ? 0.0f : NEGV;
  const float* f = feats + (long)b * TT * LL;

  for (int t = 1; t < TT; ++t) {
    float a[LL];
#pragma unroll
    for (int i = 0; i < LL; ++i) a[i] = __shfl(alpha, i, 32);
    float m = -3.402823466e38f;
#pragma unroll
    for (int i = 0; i < LL; ++i) m = fmaxf(m, tr[i] + a[i]);
    float sum = 0.0f;
#pragma unroll
    for (int i = 0; i < LL; ++i) sum += expf(tr[i] + a[i] - m);
    alpha = m + logf(sum) + f[t * LL + j];
  }

  // forward_score = logsumexp over labels of final alpha
  float av[LL];
#pragma unroll
  for (int i = 0; i < LL; ++i) av[i] = __shfl(alpha, i, 32);
  float m = -3.402823466e38f;
#pragma unroll
  for (int i = 0; i < LL; ++i) m = fmaxf(m, av[i]);
  float sum = 0.0f;
#pragma unroll
  for (int i = 0; i < LL; ++i) sum += expf(av[i] - m);
  const float fw = m + logf(sum);

  // gold score: sum_t trans[nxt,prev] + feats[b,t,nxt], t = 1..T-1
  float gold = 0.0f;
  const int* lab = labels + b * TT;
  for (int t = 1 + lane; t < TT; t += 32) {
    int prev = lab[t - 1];
    int nxt  = lab[t];
    gold += trans[nxt * LL + prev] + f[t * LL + nxt];
  }
#pragma unroll
  for (int off = 16; off > 0; off >>= 1) gold += __shfl_xor(gold, off, 32);

  if (lane == 0) per_batch[b] = fw - gold;
}

// ---------------------------------------------------------------------------
// Kernel 3: mean over 512 per-batch values -> scalar
// ---------------------------------------------------------------------------
__global__ __launch_bounds__(512) void reduce_kernel(
    const float* __restrict__ per_batch, float* __restrict__ out) {
  __shared__ float s[512];
  const int t = threadIdx.x;
  s[t] = per_batch[t];
  __syncthreads();
  for (int off = 256; off > 0; off >>= 1) {
    if (t < off) s[t] += s[t + off];
    __syncthreads();
  }
  if (t == 0) out[0] = s[0] * (1.0f / (float)BB);
}

extern "C" void kernel_launch(void* const* d_in, const int* in_sizes, int n_in,
                              void* d_out, int out_size, void* d_ws, size_t ws_size,
                              hipStream_t stream) {
  const float* X     = (const float*)d_in[0];  // (512,128,768) f32
  const float* W     = (const float*)d_in[1];  // (12,768) f32
  const float* bias  = (const float*)d_in[2];  // (12,) f32
  const float* trans = (const float*)d_in[3];  // (12,12) f32
  const int*   lab   = (const int*)d_in[4];    // (512,128) int

  float* feats     = (float*)d_ws;                       // 512*128*12 f32
  float* per_batch = feats + (size_t)BB * TT * LL;       // 512 f32

  feats_gemm_kernel<<<(BB * TT) / 64, 128, 0, stream>>>(X, W, bias, feats);
  crf_kernel<<<BB, 32, 0, stream>>>(feats, trans, lab, per_batch);
  reduce_kernel<<<1, BB, 0, stream>>>(per_batch, (float*)d_out);
}